// PairwiseEnergy_32538672234668
// MI455X (gfx1250) — compile-verified
//
#include <hip/hip_runtime.h>
#include <math.h>

typedef float v2f __attribute__((ext_vector_type(2)));
typedef float v8f __attribute__((ext_vector_type(8)));

#define D_DIM 64

// float atomic max via monotonic int-bit encoding (handles negatives)
__device__ __forceinline__ void atomicMaxFloat(float* addr, float val) {
    if (val >= 0.0f) {
        atomicMax((int*)addr, __float_as_int(val));
    } else {
        atomicMin((unsigned int*)addr, __float_as_uint(val));
    }
}

// One wave computes dot(Q2[c_e], K2[u_e]) for 16 edges via 16x16x4 f32 WMMA,
// accumulating Q_tile(16x64) x K_tile^T(64x16); edge dots are the diagonal.
// Per ISA VGPR layouts, lane l supplies row/col for edge (l&15) at
// k-offset (l>>4)*2 for BOTH the A and B fragments.
__device__ __forceinline__ bool edge_dot_wmma(
    const float* __restrict__ Q2, const float* __restrict__ K2,
    const int* __restrict__ c2, const int* __restrict__ u2,
    long long tileBase, long long E, int lane,
    float& ell_out, int& node_out)
{
    const int  eLocal = lane & 15;
    long long  edge   = tileBase + eLocal;
    long long  edgeC  = (edge < E) ? edge : (E - 1);   // clamp loads, predicate later
    const int  c = c2[edgeC];
    const int  u = u2[edgeC];
    const float* qrow = Q2 + (long long)c * D_DIM;
    const float* krow = K2 + (long long)u * D_DIM;
    const int  kband = (lane >> 4) * 2;                // lanes<16 -> K{0,1}, lanes>=16 -> K{2,3}

    v8f acc = {};
    #pragma unroll
    for (int kb = 0; kb < D_DIM; kb += 4) {
        v2f a = *(const v2f*)(qrow + kb + kband);
        v2f b = *(const v2f*)(krow + kb + kband);
        acc = __builtin_amdgcn_wmma_f32_16x16x4_f32(
            /*neg_a=*/false, a, /*neg_b=*/false, b,
            /*c_mod=*/(short)0, acc, /*reuse_a=*/false, /*reuse_b=*/false);
    }

    // Diagonal of 16x16 f32 C/D layout:
    //   lanes 0-7  : element (m=lane,   n=lane)   in acc[lane]
    //   lanes 24-31: element (m=lane-16,n=lane-16) in acc[lane-24]
    const int r = lane & 7;
    float diag = acc[0];
    if (r == 1) diag = acc[1];
    if (r == 2) diag = acc[2];
    if (r == 3) diag = acc[3];
    if (r == 4) diag = acc[4];
    if (r == 5) diag = acc[5];
    if (r == 6) diag = acc[6];
    if (r == 7) diag = acc[7];

    ell_out  = diag * 0.125f;   // 1/sqrt(64)
    node_out = c;
    return ((lane < 8) || (lane >= 24)) && (edge < E);
}

__global__ void pe_init_kernel(float* nodeMax, float* nodeSum, float* out,
                               int N, int G)
{
    int i = blockIdx.x * blockDim.x + threadIdx.x;
    if (i < N) {
        ((unsigned int*)nodeMax)[i] = 0xFF800000u;   // -inf
        nodeSum[i] = 0.0f;
    }
    if (i < G) out[i] = 0.0f;
}

// PASS 0: per-node max of ell.  PASS 1: per-node sum of exp(ell - max).
template<int PASS>
__global__ void pe_edge_pass_kernel(const float* __restrict__ Q2,
                                    const float* __restrict__ K2,
                                    const int*  __restrict__ c2,
                                    const int*  __restrict__ u2,
                                    float* __restrict__ nodeMax,
                                    float* __restrict__ nodeSum,
                                    long long E, long long numTiles)
{
    const int wavesPerBlock = blockDim.x >> 5;
    const long long tile = (long long)blockIdx.x * wavesPerBlock + (threadIdx.x >> 5);
    if (tile >= numTiles) return;               // wave-uniform: EXEC stays full for WMMA
    const int lane = threadIdx.x & 31;

    float ell; int node;
    bool p = edge_dot_wmma(Q2, K2, c2, u2, tile * 16, E, lane, ell, node);

    if (PASS == 0) {
        if (p) atomicMaxFloat(&nodeMax[node], ell);
    } else {
        if (p) {
            float m = nodeMax[node];
            atomicAdd(&nodeSum[node], __expf(ell - m));
        }
    }
}

__global__ void pe_finalize_kernel(const float* __restrict__ nodeMax,
                                   const float* __restrict__ nodeSum,
                                   const int*  __restrict__ batch,
                                   float* __restrict__ out, int N)
{
    int n = blockIdx.x * blockDim.x + threadIdx.x;
    if (n >= N) return;
    float s = nodeSum[n];
    float lse = 0.0f;
    if (s > 0.0f) {
        lse = nodeMax[n] + __logf(s);           // empty segments stay 0 (matches reference)
    }
    atomicAdd(&out[batch[n]], lse);
}

extern "C" void kernel_launch(void* const* d_in, const int* in_sizes, int n_in,
                              void* d_out, int out_size, void* d_ws, size_t ws_size,
                              hipStream_t stream)
{
    const float* Q2    = (const float*)d_in[0];
    const float* K2    = (const float*)d_in[1];
    const int*   c2    = (const int*)d_in[2];
    const int*   u2    = (const int*)d_in[3];
    const int*   batch = (const int*)d_in[4];

    const int       N = in_sizes[0] / D_DIM;
    const long long E = (long long)in_sizes[2];
    const int       G = out_size;

    float* nodeMax = (float*)d_ws;          // N floats
    float* nodeSum = nodeMax + N;           // N floats
    float* out     = (float*)d_out;         // G floats

    // init scratch + output
    {
        int m = (N > G) ? N : G;
        int blocks = (m + 255) / 256;
        pe_init_kernel<<<dim3(blocks), dim3(256), 0, stream>>>(nodeMax, nodeSum, out, N, G);
    }

    const long long numTiles = (E + 15) / 16;
    const int  block = 256;                                  // 8 waves / block
    const int  wavesPerBlock = block / 32;
    const long long blocks = (numTiles + wavesPerBlock - 1) / wavesPerBlock;

    pe_edge_pass_kernel<0><<<dim3((unsigned int)blocks), dim3(block), 0, stream>>>(
        Q2, K2, c2, u2, nodeMax, nodeSum, E, numTiles);
    pe_edge_pass_kernel<1><<<dim3((unsigned int)blocks), dim3(block), 0, stream>>>(
        Q2, K2, c2, u2, nodeMax, nodeSum, E, numTiles);

    pe_finalize_kernel<<<dim3((N + 255) / 256), dim3(256), 0, stream>>>(
        nodeMax, nodeSum, batch, out, N);
}